// PredictionHead_38482906972422
// MI455X (gfx1250) — compile-verified
//
#include <hip/hip_runtime.h>
#include <hip/hip_bf16.h>

typedef __attribute__((ext_vector_type(2))) float v2f;
typedef __attribute__((ext_vector_type(8))) float v8f;

#define HDIM 768
#define BATCH 256

// -------------------------------------------------------------------------
// Zero-fill a float buffer (grid-stride).
// -------------------------------------------------------------------------
__global__ void zero_kernel(float* __restrict__ p, int n) {
    int i = blockIdx.x * blockDim.x + threadIdx.x;
    int stride = gridDim.x * blockDim.x;
    for (; i < n; i += stride) p[i] = 0.0f;
}

// -------------------------------------------------------------------------
// Sorted-segment sum: feat [nRows, 768] f32, seg [nRows] sorted int32.
// Accumulates into out[seg, colOff : colOff+768] with row stride ldo.
// Block = 192 threads, each owning 4 consecutive columns (float4 / b128
// coalesced loads). Rows are walked sequentially; the per-thread accumulator
// is flushed with atomicAdd only when the (uniform, scalar-loaded) segment id
// changes -> atomics are negligible vs the 1.84 GB streaming read.
// -------------------------------------------------------------------------
__global__ void segsum_kernel(const float* __restrict__ feat,
                              const int* __restrict__ seg,
                              float* __restrict__ out, int ldo, int colOff,
                              int nRows, int rowsPer) {
    const int c4 = threadIdx.x * 4;                 // 0..764
    int r0 = blockIdx.x * rowsPer;
    int r1 = r0 + rowsPer;
    if (r1 > nRows) r1 = nRows;
    if (r0 >= r1) return;

    float4 acc = make_float4(0.f, 0.f, 0.f, 0.f);
    int cur = seg[r0];                              // uniform -> s_load
    for (int r = r0; r < r1; ++r) {
        int s = seg[r];                             // uniform -> s_load
        if (s != cur) {
            float* o = out + (size_t)cur * ldo + colOff + c4;
            atomicAdd(o + 0, acc.x); atomicAdd(o + 1, acc.y);
            atomicAdd(o + 2, acc.z); atomicAdd(o + 3, acc.w);
            acc = make_float4(0.f, 0.f, 0.f, 0.f);
            cur = s;
        }
        const float4 v = *(const float4*)(feat + (size_t)r * HDIM + c4);
        acc.x += v.x; acc.y += v.y; acc.z += v.z; acc.w += v.w;
    }
    float* o = out + (size_t)cur * ldo + colOff + c4;
    atomicAdd(o + 0, acc.x); atomicAdd(o + 1, acc.y);
    atomicAdd(o + 2, acc.z); atomicAdd(o + 3, acc.w);
}

// -------------------------------------------------------------------------
// FP32 WMMA GEMM:  C[M,N] = op(A[M,K] @ W[K,N] + bias),  all row-major.
// One wave computes a 64x16 output tile: 4 M-subtiles share one B fragment,
// so each K-step of 4 issues 4 A b64-loads + 2 B b32-loads feeding 4
// independent back-to-back V_WMMA_F32_16X16X4_F32 (no D->A/B RAW hazard,
// full XDL pipelining, exact fp32 like the reference).
//
// Fragment layouts per CDNA5 ISA 7.12.2:
//   A (16x4):  lane l -> row m=l&15; VGPR0 = A[m][k0+(l>>4)*2], VGPR1 = +1
//   B (4x16):  lane l -> col n=l&15; VGPR0 = W[k0+(l>>4)*2][n], VGPR1 = +1
//   C/D:       VGPR v, lane l -> row v+(l>>4)*8, col l&15
//
// Ragged N (3129): loads use a CLAMPED column (min(n, N-1)) so the hot loop
// has no exec-mask manipulation; garbage lands only in lanes whose columns
// fail the store-side bound check (columns are lane-mapped in D).
// -------------------------------------------------------------------------
__global__ __launch_bounds__(32)
void gemm_f32_wmma(const float* __restrict__ A, int lda,
                   const float* __restrict__ W, int ldw,
                   const float* __restrict__ bias,
                   float* __restrict__ C, int ldc,
                   int N, int K, int doRelu) {
    const int lane  = threadIdx.x;          // 0..31
    const int m0    = blockIdx.y * 64;      // 4 x 16 M-subtiles per wave
    const int n0    = blockIdx.x * 16;
    const int mrow  = lane & 15;
    const int ncol  = lane & 15;
    const int kpair = (lane >> 4) * 2;      // 0 or 2
    const bool nOk  = (n0 + ncol) < N;
    const int n     = nOk ? (n0 + ncol) : (N - 1);   // clamped, loop is guard-free

    const float* a0 = A + (size_t)(m0 + 0  + mrow) * lda + kpair;
    const float* a1 = A + (size_t)(m0 + 16 + mrow) * lda + kpair;
    const float* a2 = A + (size_t)(m0 + 32 + mrow) * lda + kpair;
    const float* a3 = A + (size_t)(m0 + 48 + mrow) * lda + kpair;
    const float* w0 = W + (size_t)kpair * ldw + n;

    v8f acc0 = {}, acc1 = {}, acc2 = {}, acc3 = {};
    for (int k0 = 0; k0 < K; k0 += 4) {
        v2f b;
        b.x = w0[(size_t)k0 * ldw];
        b.y = w0[(size_t)(k0 + 1) * ldw];
        v2f fa0, fa1, fa2, fa3;
        fa0.x = a0[k0]; fa0.y = a0[k0 + 1];
        fa1.x = a1[k0]; fa1.y = a1[k0 + 1];
        fa2.x = a2[k0]; fa2.y = a2[k0 + 1];
        fa3.x = a3[k0]; fa3.y = a3[k0 + 1];
        acc0 = __builtin_amdgcn_wmma_f32_16x16x4_f32(false, fa0, false, b, (short)0, acc0, false, false);
        acc1 = __builtin_amdgcn_wmma_f32_16x16x4_f32(false, fa1, false, b, (short)0, acc1, false, false);
        acc2 = __builtin_amdgcn_wmma_f32_16x16x4_f32(false, fa2, false, b, (short)0, acc2, false, false);
        acc3 = __builtin_amdgcn_wmma_f32_16x16x4_f32(false, fa3, false, b, (short)0, acc3, false, false);
    }

    if (nOk) {
        const float bv = bias[n];
        const int mhalf = (lane >> 4) * 8;
        #pragma unroll
        for (int v = 0; v < 8; ++v) {
            float x0 = acc0[v] + bv;
            float x1 = acc1[v] + bv;
            float x2 = acc2[v] + bv;
            float x3 = acc3[v] + bv;
            if (doRelu) {
                x0 = fmaxf(x0, 0.f); x1 = fmaxf(x1, 0.f);
                x2 = fmaxf(x2, 0.f); x3 = fmaxf(x3, 0.f);
            }
            C[(size_t)(m0 + 0  + mhalf + v) * ldc + n] = x0;
            C[(size_t)(m0 + 16 + mhalf + v) * ldc + n] = x1;
            C[(size_t)(m0 + 32 + mhalf + v) * ldc + n] = x2;
            C[(size_t)(m0 + 48 + mhalf + v) * ldc + n] = x3;
        }
    }
}

// -------------------------------------------------------------------------
// Launch: zero scratch -> segsum(node) -> segsum(edge) -> 4 WMMA GEMMs.
// -------------------------------------------------------------------------
extern "C" void kernel_launch(void* const* d_in, const int* in_sizes, int n_in,
                              void* d_out, int out_size, void* d_ws, size_t ws_size,
                              hipStream_t stream) {
    const float* node_feat    = (const float*)d_in[0];
    const float* edge_feat    = (const float*)d_in[1];
    const float* question_emb = (const float*)d_in[2];
    const int*   node_seg     = (const int*)d_in[3];
    const int*   edge_seg     = (const int*)d_in[4];
    const float* W_lang       = (const float*)d_in[5];
    const float* b_lang       = (const float*)d_in[6];
    const float* W_mid        = (const float*)d_in[7];
    const float* b_mid        = (const float*)d_in[8];
    const float* W_mid2       = (const float*)d_in[9];
    const float* b_mid2       = (const float*)d_in[10];
    const float* W_ans        = (const float*)d_in[11];
    const float* b_ans        = (const float*)d_in[12];
    float*       pred         = (float*)d_out;

    const int nNodes = in_sizes[0] / HDIM;   // 300000
    const int nEdges = in_sizes[1] / HDIM;   // 300000
    const int A      = in_sizes[12];         // 3129

    // Workspace layout (floats): lang_g [256,2304] | mid1 [256,768] | mid2 [256,768]
    float* langg = (float*)d_ws;
    float* mid1  = langg + (size_t)BATCH * 3 * HDIM;
    float* mid2  = mid1 + (size_t)BATCH * HDIM;

    // 1) zero lang_g (segment sums accumulate into cols [0, 1536))
    {
        int total = BATCH * 3 * HDIM;
        zero_kernel<<<512, 256, 0, stream>>>(langg, total);
    }

    // 2) sorted segment sums -> lang_g[:, 0:768] (nodes), [:, 768:1536] (edges)
    {
        const int chunks = 2048;
        int rowsPerN = (nNodes + chunks - 1) / chunks;
        int rowsPerE = (nEdges + chunks - 1) / chunks;
        segsum_kernel<<<chunks, 192, 0, stream>>>(node_feat, node_seg, langg,
                                                  3 * HDIM, 0, nNodes, rowsPerN);
        segsum_kernel<<<chunks, 192, 0, stream>>>(edge_feat, edge_seg, langg,
                                                  3 * HDIM, HDIM, nEdges, rowsPerE);
    }

    // 3) lang = Q @ W_lang + b_lang  -> lang_g[:, 1536:2304]
    gemm_f32_wmma<<<dim3(HDIM / 16, BATCH / 64), 32, 0, stream>>>(
        question_emb, HDIM, W_lang, HDIM, b_lang,
        langg + 2 * HDIM, 3 * HDIM, HDIM, HDIM, /*relu=*/0);

    // 4) mid1 = relu(lang_g @ W_mid + b_mid)
    gemm_f32_wmma<<<dim3(HDIM / 16, BATCH / 64), 32, 0, stream>>>(
        langg, 3 * HDIM, W_mid, HDIM, b_mid,
        mid1, HDIM, HDIM, 3 * HDIM, /*relu=*/1);

    // 5) mid2 = relu(mid1 @ W_mid2 + b_mid2)
    gemm_f32_wmma<<<dim3(HDIM / 16, BATCH / 64), 32, 0, stream>>>(
        mid1, HDIM, W_mid2, HDIM, b_mid2,
        mid2, HDIM, HDIM, HDIM, /*relu=*/1);

    // 6) pred = mid2 @ W_ans + b_ans  (N = 3129, ragged tail via clamped loads)
    gemm_f32_wmma<<<dim3((A + 15) / 16, BATCH / 64), 32, 0, stream>>>(
        mid2, HDIM, W_ans, A, b_ans,
        pred, A, A, HDIM, /*relu=*/0);
}